// BILSTM_CRF_Span_Attr_Tail_31748398252325
// MI455X (gfx1250) — compile-verified
//
#include <hip/hip_runtime.h>
#include <hip/hip_bf16.h>

// ---------------------------------------------------------------------------
// BILSTM-CRF span/attr tail.
//   logits_span = span_hiddens @ W_span + b_span
//   logits_attr = scatter_mean(attr_hiddens) @ W_attr + b_attr
// Key identity: segment-mean commutes with the linear layer, so we run a
// branch-free dual streaming GEMM (the 200 MB bandwidth-critical part) and
// apply the scatter-mean on the tiny [B,S,13] logits afterwards.
// f32 WMMA 16x16x4; B-fragments staged once per block in LDS.
// ---------------------------------------------------------------------------

typedef __attribute__((ext_vector_type(2))) float v2f;
typedef __attribute__((ext_vector_type(8))) float v8f;

#define BN 64
#define SEQ 512
#define HID 768
#define NSPAN 5
#define NATTR 13
#define KSTEPS (HID / 4)          // 192
#define FRAG_ELEMS (KSTEPS * 64)  // 12288 floats per packed weight matrix
#define B_ID 1
#define E_ID 3
#define WAVES_PER_BLOCK 8

// ---------------------------------------------------------------------------
// Kernel 1: start[b][s] = max(last B index <= s, 0)
// ---------------------------------------------------------------------------
__global__ void scan_starts_kernel(const int* __restrict__ labels,
                                   int* __restrict__ start) {
    const int b = blockIdx.x;
    if (threadIdx.x != 0) return;
    const int* L = labels + b * SEQ;
    int* st = start + b * SEQ;
    int last = 0;
    for (int s = 0; s < SEQ; ++s) {
        if (L[s] == B_ID) last = s;
        st[s] = last;
    }
}

// ---------------------------------------------------------------------------
// Kernel 2: pack W (zero-padded N->16) into the 16x16x4 B-fragment layout:
//   frag[t][lane][e] = W[4t + 2*(lane>>4) + e][lane&15]
// ---------------------------------------------------------------------------
__global__ void pack_weights_kernel(const float* __restrict__ Wsp,
                                    const float* __restrict__ Wat,
                                    float* __restrict__ fsp,
                                    float* __restrict__ fat) {
    const int i = blockIdx.x * blockDim.x + threadIdx.x;
    if (i >= FRAG_ELEMS) return;
    const int e    = i & 1;
    const int lane = (i >> 1) & 31;
    const int t    = i >> 6;
    const int k    = 4 * t + 2 * (lane >> 4) + e;
    const int n    = lane & 15;
    fsp[i] = (n < NSPAN) ? Wsp[k * NSPAN + n] : 0.0f;
    fat[i] = (n < NATTR) ? Wat[k * NATTR + n] : 0.0f;
}

// ---------------------------------------------------------------------------
// Kernel 3: branch-free dual GEMM. 8 waves/block, each wave owns one 16-row
// tile. Both packed B-fragment arrays (96 KB) staged in LDS once per block.
// ---------------------------------------------------------------------------
extern __shared__ float smem[];

__global__ __launch_bounds__(WAVES_PER_BLOCK * 32) void dual_gemm_kernel(
    const float* __restrict__ spanH,
    const float* __restrict__ attrH,
    const float* __restrict__ fsp,
    const float* __restrict__ fat,
    const float* __restrict__ bsp,
    const float* __restrict__ bat,
    float* __restrict__ out_span,
    float* __restrict__ out_attr) {

    float* fragS = smem;                // FRAG_ELEMS floats
    float* fragA = smem + FRAG_ELEMS;   // FRAG_ELEMS floats

    // cooperative one-time preload of both packed weight fragments (16B chunks)
    {
        const int tid = threadIdx.x;
        const float4* gs = (const float4*)fsp;
        const float4* ga = (const float4*)fat;
        float4* ls = (float4*)fragS;
        float4* la = (float4*)fragA;
        for (int i = tid; i < FRAG_ELEMS / 4; i += WAVES_PER_BLOCK * 32) {
            ls[i] = gs[i];
            la[i] = ga[i];
        }
    }
    __syncthreads();

    const int wave = threadIdx.x >> 5;
    const int lane = threadIdx.x & 31;
    const int r    = lane & 15;
    const int hi   = lane >> 4;

    // 4 blocks per batch, 8 tiles (128 rows) per block
    const int b    = blockIdx.x >> 2;
    const int s0   = ((blockIdx.x & 3) * WAVES_PER_BLOCK + wave) * 16;
    const int s    = s0 + r;
    const int row  = b * SEQ + s;

    // lane's A stream: row (lane&15), k-pair base offset 2*hi
    const float* spBase = spanH + row * HID + 2 * hi;
    const float* atBase = attrH + row * HID + 2 * hi;

    v8f accS = {};
    v8f accA = {};

    for (int tt = 0; tt < KSTEPS / 8; ++tt) {
        // each outer step consumes one 128B line per row; prefetch 2 ahead
        __builtin_prefetch(spBase + 32 * tt + 64, 0, 3);
        __builtin_prefetch(atBase + 32 * tt + 64, 0, 3);
#pragma unroll
        for (int u = 0; u < 8; ++u) {
            const int t    = tt * 8 + u;
            const int koff = 4 * t;
            v2f aS = *(const v2f*)(spBase + koff);
            v2f aA = *(const v2f*)(atBase + koff);
            v2f bS = *(const v2f*)(fragS + t * 64 + lane * 2);
            v2f bA = *(const v2f*)(fragA + t * 64 + lane * 2);
            accS = __builtin_amdgcn_wmma_f32_16x16x4_f32(
                false, aS, false, bS, (short)0, accS, false, false);
            accA = __builtin_amdgcn_wmma_f32_16x16x4_f32(
                false, aA, false, bA, (short)0, accA, false, false);
        }
    }

    // C/D layout: VGPR v, lanes 0-15 -> M=v, lanes 16-31 -> M=v+8; N = lane&15
    const int n = r;
#pragma unroll
    for (int v = 0; v < 8; ++v) {
        const int m    = v + 8 * hi;
        const int orow = b * SEQ + s0 + m;
        if (n < NSPAN) out_span[orow * NSPAN + n] = accS[v] + bsp[n];
        if (n < NATTR) out_attr[orow * NATTR + n] = accA[v] + bat[n];
    }
}

// ---------------------------------------------------------------------------
// Kernel 4: for E rows, mean of logits_attr rows [start..s] -> meanbuf.
// (bias & projection commute with the mean). Reads only pre-scatter values.
// ---------------------------------------------------------------------------
__global__ void attr_mean_kernel(const int* __restrict__ labels,
                                 const int* __restrict__ start,
                                 const float* __restrict__ out_attr,
                                 float* __restrict__ meanbuf) {
    const int idx = blockIdx.x * blockDim.x + threadIdx.x;  // b*SEQ + s
    if (idx >= BN * SEQ) return;
    if (labels[idx] != E_ID) return;
    const int b   = idx / SEQ;
    const int s   = idx - b * SEQ;
    const int st  = start[idx];
    const int cnt = s - st + 1;
    const float inv = 1.0f / (float)cnt;
    const float* p = out_attr + (idx - (s - st)) * NATTR;  // row (b*SEQ+st)
    float acc[NATTR];
#pragma unroll
    for (int n = 0; n < NATTR; ++n) acc[n] = 0.0f;
    for (int j = 0; j < cnt; ++j)
#pragma unroll
        for (int n = 0; n < NATTR; ++n) acc[n] += p[j * NATTR + n];
    float* o = meanbuf + idx * NATTR;
#pragma unroll
    for (int n = 0; n < NATTR; ++n) o[n] = acc[n] * inv;
}

// ---------------------------------------------------------------------------
// Kernel 5: scatter the means into logits_attr at E rows.
// ---------------------------------------------------------------------------
__global__ void attr_scatter_kernel(const int* __restrict__ labels,
                                    const float* __restrict__ meanbuf,
                                    float* __restrict__ out_attr) {
    const int idx = blockIdx.x * blockDim.x + threadIdx.x;  // over BN*SEQ*NATTR
    if (idx >= BN * SEQ * NATTR) return;
    const int row = idx / NATTR;
    if (labels[row] == E_ID) out_attr[idx] = meanbuf[idx];
}

// ---------------------------------------------------------------------------
extern "C" void kernel_launch(void* const* d_in, const int* in_sizes, int n_in,
                              void* d_out, int out_size, void* d_ws, size_t ws_size,
                              hipStream_t stream) {
    const int*   labels = (const int*)d_in[0];
    const float* spanH  = (const float*)d_in[1];
    const float* attrH  = (const float*)d_in[2];
    const float* Wsp    = (const float*)d_in[3];
    const float* bsp    = (const float*)d_in[4];
    const float* Wat    = (const float*)d_in[5];
    const float* bat    = (const float*)d_in[6];

    float* out_span = (float*)d_out;
    float* out_attr = out_span + (size_t)BN * SEQ * NSPAN;

    // workspace layout (16B aligned sections)
    char* ws = (char*)d_ws;
    int*   start   = (int*)ws;                                    // 128 KB
    float* fsp     = (float*)(ws + (size_t)BN * SEQ * 4);         // 48 KB
    float* fat     = fsp + FRAG_ELEMS;                            // 48 KB
    float* meanbuf = fat + FRAG_ELEMS;                            // 1.7 MB

    scan_starts_kernel<<<BN, 32, 0, stream>>>(labels, start);

    pack_weights_kernel<<<(FRAG_ELEMS + 255) / 256, 256, 0, stream>>>(Wsp, Wat,
                                                                      fsp, fat);

    {
        const int grid  = BN * 4;                       // 8 tiles per block
        const int block = WAVES_PER_BLOCK * 32;         // 256
        const size_t lds = 2 * FRAG_ELEMS * sizeof(float);  // 96 KB
        dual_gemm_kernel<<<grid, block, lds, stream>>>(spanH, attrH, fsp, fat,
                                                       bsp, bat,
                                                       out_span, out_attr);
    }

    attr_mean_kernel<<<(BN * SEQ + 255) / 256, 256, 0, stream>>>(labels, start,
                                                                 out_attr, meanbuf);

    attr_scatter_kernel<<<(BN * SEQ * NATTR + 255) / 256, 256, 0, stream>>>(
        labels, meanbuf, out_attr);
}